// GetACEDensity_26783416058560
// MI455X (gfx1250) — compile-verified
//
#include <hip/hip_runtime.h>

// ---------------- problem constants (match reference) ----------------
#define NBATCH 32
#define NATOM  256
#define PAIRS  15360
#define NWAVE  8
#define NORBIT 128
#define NO     13                 // 1 + 3 + 9 angular channels
#define NTYPE  4
#define SUMST  (NO * NWAVE)       // 104 floats per atom accumulator
#define BATCH_SUM (NATOM * SUMST) // 26624 floats = 104 KB per batch
#define HYP_F  (3 * NWAVE * NORBIT) // 3072 floats
#define TAB_F  (3 * NTYPE * NWAVE)  // rs | inta | params = 96 floats
#define SLICES 8
#define TPB    256

typedef float v2f __attribute__((ext_vector_type(2)));
typedef float v8f __attribute__((ext_vector_type(8)));

// Raw LDS byte offset of a generic pointer that points into shared memory.
__device__ __forceinline__ unsigned lds_offset(const void* p) {
    return (unsigned)(uintptr_t)(__attribute__((address_space(3))) const void*)p;
}

// Async copy of `n16` 16-byte chunks (thread-strided) from global to LDS.
// Uses CDNA5 GLOBAL_LOAD_ASYNC_TO_LDS_B128 (ASYNCcnt-tracked, bypasses VGPRs).
__device__ __forceinline__ void async_g2l_b128(float* dstLds, const float* srcGlb,
                                               int n16, int tid) {
    const unsigned lds0 = lds_offset(dstLds);
    for (int c = tid; c < n16; c += TPB) {
        const unsigned lds = lds0 + (unsigned)c * 16u;
        const float* g = srcGlb + (size_t)c * 4;
        asm volatile("global_load_async_to_lds_b128 %0, %1, off"
                     :: "v"(lds), "v"(g) : "memory");
    }
}

__device__ __forceinline__ void wait_async0() {
    asm volatile("s_wait_asynccnt 0" ::: "memory");
}

// ---------------- pair phase: radial*angular -> LDS scatter-add ----------------
__device__ __forceinline__ void pair_accum(
    const float* __restrict__ cart, const int* __restrict__ species,
    const int* __restrict__ atom_index, const float* __restrict__ shifts,
    const float* __restrict__ tab /* LDS: rs|inta|params */,
    float* __restrict__ sh_sum, int b, int p)
{
    // atom_index layout: (2, NBATCH, PAIRS)
    const int i0 = atom_index[(size_t)b * PAIRS + p];
    const int i1 = atom_index[(size_t)(NBATCH + b) * PAIRS + p];
    const float* c0 = cart + ((size_t)b * NATOM + i0) * 3;
    const float* c1 = cart + ((size_t)b * NATOM + i1) * 3;
    const float* sh = shifts + ((size_t)b * PAIRS + p) * 3;

    float dx = c0[0] - c1[0] + sh[0];
    float dy = c0[1] - c1[1] + sh[1];
    float dz = c0[2] - c1[2] + sh[2];
    float r  = sqrtf(dx * dx + dy * dy + dz * dz);
    float inv = 1.0f / r;
    float ux = dx * inv, uy = dy * inv, uz = dz * inv;

    const int sp = species[b * NATOM + i1];
    const float* rsw = tab + sp * NWAVE;
    const float* iaw = tab + NTYPE * NWAVE + sp * NWAVE;
    const float* pmw = tab + 2 * NTYPE * NWAVE + sp * NWAVE;

    float rad[NWAVE];
#pragma unroll
    for (int w = 0; w < NWAVE; ++w) {
        float t = r - rsw[w];
        rad[w] = __expf(iaw[w] * t * t) * pmw[w];  // fold orb_coeff (params[sp])
    }
    const float ang[NO] = {1.0f, ux, uy, uz,
                           ux * ux, ux * uy, ux * uz,
                           uy * ux, uy * uy, uy * uz,
                           uz * ux, uz * uy, uz * uz};

    float* base = sh_sum + i0 * SUMST;
#pragma unroll
    for (int a = 0; a < NO; ++a) {
#pragma unroll
        for (int w = 0; w < NWAVE; ++w)
            atomicAdd(&base[a * NWAVE + w], ang[a] * rad[w]);  // ds_add_f32
    }
}

// ---------------- contraction phase: per-channel GEMM via V_WMMA_F32_16X16X4_F32 ----------------
// For each angular channel a: G_a = S_a(256x8) @ H_a(8x128); density += G_a^2 (elementwise).
// A 16x4 f32 layout: lane L holds row M=L%16; VGPR v holds K = 2*(L/16) + v.
// C/D 16x16 f32 layout: lane L holds col N=L%16; VGPR r holds row M = r + 8*(L/16).
__device__ __forceinline__ void density_wmma(
    const float* __restrict__ sh_sum, const float* __restrict__ sh_hyp,
    float* __restrict__ out, int b)
{
    const int tid    = threadIdx.x;
    const int wave   = tid >> 5;
    const int lane   = tid & 31;
    const int laneHi = lane >> 4;   // 0 or 1
    const int laneLo = lane & 15;

    for (int mt = wave * 2; mt < wave * 2 + 2; ++mt) {       // 16 M-tiles over 8 waves
        const int rowA = mt * 16 + laneLo;                   // A-matrix row for this lane
        for (int nt = 0; nt < NORBIT / 16; ++nt) {           // 8 N-tiles
            const int col = nt * 16 + laneLo;
            v8f dacc = {0.f, 0.f, 0.f, 0.f, 0.f, 0.f, 0.f, 0.f};
            for (int a = 0; a < NO; ++a) {
                const int lvl = (a == 0) ? 0 : ((a < 4) ? 1 : 2);  // INDEX_PARA
                const float* S = sh_sum + rowA * SUMST + a * NWAVE;
                const float* H = sh_hyp + lvl * NWAVE * NORBIT;
                v8f c = {0.f, 0.f, 0.f, 0.f, 0.f, 0.f, 0.f, 0.f};
#pragma unroll
                for (int j = 0; j < 2; ++j) {                // K=8 as two K=4 WMMAs
                    const int k0 = 4 * j + 2 * laneHi;
                    v2f av; av.x = S[k0];                 av.y = S[k0 + 1];
                    v2f bv; bv.x = H[k0 * NORBIT + col];  bv.y = H[(k0 + 1) * NORBIT + col];
                    c = __builtin_amdgcn_wmma_f32_16x16x4_f32(
                            false, av, false, bv, (short)0, c, false, false);
                }
#pragma unroll
                for (int r = 0; r < 8; ++r) dacc[r] += c[r] * c[r];  // square before channel-sum
            }
#pragma unroll
            for (int r = 0; r < 8; ++r) {
                const int atomRow = mt * 16 + r + laneHi * 8;
                out[((size_t)b * NATOM + atomRow) * NORBIT + nt * 16 + laneLo] = dacc[r];
            }
        }
    }
}

// ---------------- kernels ----------------
__global__ void __launch_bounds__(TPB)
ace_pairs_kernel(const float* __restrict__ cart, const int* __restrict__ species,
                 const int* __restrict__ atom_index, const float* __restrict__ shifts,
                 const float* __restrict__ rs, const float* __restrict__ inta,
                 const float* __restrict__ params, float* __restrict__ gsum)
{
    extern __shared__ float smem[];
    float* sh_sum = smem;              // BATCH_SUM floats
    float* sh_tab = smem + BATCH_SUM;  // TAB_F floats

    const int b   = blockIdx.x / SLICES;
    const int sl  = blockIdx.x % SLICES;
    const int tid = threadIdx.x;

    for (int i = tid; i < BATCH_SUM; i += TPB) sh_sum[i] = 0.f;
    if (tid < NTYPE * NWAVE) {
        sh_tab[tid]                    = rs[tid];
        sh_tab[NTYPE * NWAVE + tid]    = inta[tid];
        sh_tab[2 * NTYPE * NWAVE + tid]= params[tid];
    }
    __syncthreads();

    const int per = PAIRS / SLICES;  // 1920 pairs per slice
    for (int p = sl * per + tid; p < (sl + 1) * per; p += TPB)
        pair_accum(cart, species, atom_index, shifts, sh_tab, sh_sum, b, p);
    __syncthreads();

    float* g = gsum + (size_t)b * BATCH_SUM;
    for (int i = tid; i < BATCH_SUM; i += TPB) {
        float v = sh_sum[i];
        if (v != 0.f) unsafeAtomicAdd(&g[i], v);  // global_atomic_add_f32
    }
}

__global__ void __launch_bounds__(TPB)
ace_density_kernel(const float* __restrict__ gsum, const float* __restrict__ hyper,
                   float* __restrict__ out)
{
    extern __shared__ float smem[];
    float* sh_sum = smem;               // BATCH_SUM floats
    float* sh_hyp = smem + BATCH_SUM;   // HYP_F floats
    const int b = blockIdx.x, tid = threadIdx.x;

    // CDNA5 async global->LDS copy (no VGPR staging), ASYNCcnt-tracked.
    async_g2l_b128(sh_sum, gsum + (size_t)b * BATCH_SUM, BATCH_SUM / 4, tid); // 26 ops/thread
    async_g2l_b128(sh_hyp, hyper, HYP_F / 4, tid);                            // 3 ops/thread
    wait_async0();
    __syncthreads();

    density_wmma(sh_sum, sh_hyp, out, b);
}

// Fallback if d_ws is too small: fully fused, one block per batch.
__global__ void __launch_bounds__(TPB)
ace_fused_kernel(const float* __restrict__ cart, const int* __restrict__ species,
                 const int* __restrict__ atom_index, const float* __restrict__ shifts,
                 const float* __restrict__ rs, const float* __restrict__ inta,
                 const float* __restrict__ params, const float* __restrict__ hyper,
                 float* __restrict__ out)
{
    extern __shared__ float smem[];
    float* sh_sum = smem;
    float* sh_hyp = smem + BATCH_SUM;
    float* sh_tab = sh_hyp + HYP_F;
    const int b = blockIdx.x, tid = threadIdx.x;

    // Kick off the hyper-parameter copy asynchronously; it overlaps the whole
    // pair-accumulation phase and is only waited on before the WMMA phase.
    async_g2l_b128(sh_hyp, hyper, HYP_F / 4, tid);

    for (int i = tid; i < BATCH_SUM; i += TPB) sh_sum[i] = 0.f;
    if (tid < NTYPE * NWAVE) {
        sh_tab[tid]                     = rs[tid];
        sh_tab[NTYPE * NWAVE + tid]     = inta[tid];
        sh_tab[2 * NTYPE * NWAVE + tid] = params[tid];
    }
    __syncthreads();

    for (int p = tid; p < PAIRS; p += TPB)
        pair_accum(cart, species, atom_index, shifts, sh_tab, sh_sum, b, p);

    wait_async0();       // hyper copy complete (per-wave), then block barrier
    __syncthreads();

    density_wmma(sh_sum, sh_hyp, out, b);
}

// ---------------- host launcher ----------------
extern "C" void kernel_launch(void* const* d_in, const int* in_sizes, int n_in,
                              void* d_out, int out_size, void* d_ws, size_t ws_size,
                              hipStream_t stream)
{
    const float* cart   = (const float*)d_in[0];
    // d_in[1] = numatoms (all == NATOM, unused)
    const int*   spec   = (const int*)d_in[2];
    const int*   aidx   = (const int*)d_in[3];
    const float* shifts = (const float*)d_in[4];
    const float* rs     = (const float*)d_in[5];
    const float* inta   = (const float*)d_in[6];
    const float* params = (const float*)d_in[7];
    const float* hyper  = (const float*)d_in[8];
    float* out = (float*)d_out;
    (void)in_sizes; (void)n_in; (void)out_size;

    const size_t sum_bytes  = (size_t)NBATCH * BATCH_SUM * sizeof(float);
    const size_t smem_pairs = (size_t)(BATCH_SUM + TAB_F) * sizeof(float);          // ~105 KB
    const size_t smem_dens  = (size_t)(BATCH_SUM + HYP_F) * sizeof(float);          // ~116 KB
    const size_t smem_fused = (size_t)(BATCH_SUM + HYP_F + TAB_F) * sizeof(float);  // ~117 KB

    hipFuncSetAttribute((const void*)ace_pairs_kernel,
                        hipFuncAttributeMaxDynamicSharedMemorySize, (int)smem_pairs);
    hipFuncSetAttribute((const void*)ace_density_kernel,
                        hipFuncAttributeMaxDynamicSharedMemorySize, (int)smem_dens);
    hipFuncSetAttribute((const void*)ace_fused_kernel,
                        hipFuncAttributeMaxDynamicSharedMemorySize, (int)smem_fused);

    if (ws_size >= sum_bytes) {
        hipMemsetAsync(d_ws, 0, sum_bytes, stream);
        ace_pairs_kernel<<<NBATCH * SLICES, TPB, smem_pairs, stream>>>(
            cart, spec, aidx, shifts, rs, inta, params, (float*)d_ws);
        ace_density_kernel<<<NBATCH, TPB, smem_dens, stream>>>(
            (const float*)d_ws, hyper, out);
    } else {
        ace_fused_kernel<<<NBATCH, TPB, smem_fused, stream>>>(
            cart, spec, aidx, shifts, rs, inta, params, hyper, out);
    }
}